// GestureRecognitionModel_22333829939859
// MI455X (gfx1250) — compile-verified
//
#include <hip/hip_runtime.h>

// LSTM: T=4096, B=512, NIN=42, NH=9 (gates G = 4*NH = 36)
#define TT   4096
#define BB   512
#define NIN  42
#define NH   9
#define G    36      // 4*NH
#define KP   44      // K padded to multiple of 4 for 16x16x4 WMMA
#define NPT  48      // N padded to 3 WMMA tiles of 16

typedef __attribute__((ext_vector_type(2))) float v2f;
typedef __attribute__((ext_vector_type(8))) float v8f;

// ---------------------------------------------------------------------------
// Phase 1: x_pre[t,b,:] = x[t,b,:] @ W_ih^T + (b_ih + b_hh)
// M = T*B = 2,097,152 rows (131072 tiles of 16), N = 36 (3 tiles), K = 42->44.
// One wave per 16-row M-tile, 8 waves per block. Memory-bound; WMMA f32.
// ---------------------------------------------------------------------------
__global__ __launch_bounds__(256) void lstm_xproj_wmma(
    const float* __restrict__ x, const float* __restrict__ W_ih,
    const float* __restrict__ b_ih, const float* __restrict__ b_hh,
    float* __restrict__ xpre)
{
    // B matrix (K x N) staged in LDS with (K-pair, N) layout -> aligned b64 reads
    __shared__ float Bp[KP / 2][NPT][2];
    __shared__ float bsum[NPT];

    const int tid = threadIdx.x;
    for (int idx = tid; idx < (KP / 2) * NPT; idx += 256) {
        int kp = idx / NPT;
        int n  = idx % NPT;
        int k0 = kp * 2, k1 = kp * 2 + 1;
        Bp[kp][n][0] = (k0 < NIN && n < G) ? W_ih[n * NIN + k0] : 0.0f;
        Bp[kp][n][1] = (k1 < NIN && n < G) ? W_ih[n * NIN + k1] : 0.0f;
    }
    for (int n = tid; n < NPT; n += 256)
        bsum[n] = (n < G) ? (b_ih[n] + b_hh[n]) : 0.0f;
    __syncthreads();

    const int lane = tid & 31;
    const int wave = tid >> 5;
    const int tile = blockIdx.x * 8 + wave;         // [0, 131072)
    const long row0 = (long)tile * 16;

    const int m  = lane & 15;                       // A row within tile
    const int kh = lane >> 4;                       // K-half select
    const int nl = lane & 15;                       // B/C column within tile
    const float* __restrict__ xr = x + (row0 + m) * (long)NIN;

    v8f c0 = {}; v8f c1 = {}; v8f c2 = {};

#pragma unroll
    for (int kk = 0; kk < KP / 4; ++kk) {
        const int k0 = kk * 4 + 2 * kh;             // this lane's K base
        v2f a;
        if (k0 + 1 < NIN) {                         // only kk==10, kh==1 is padded
            a[0] = xr[k0];
            a[1] = xr[k0 + 1];
        } else {
            a[0] = 0.0f; a[1] = 0.0f;
        }
        const int kp = k0 >> 1;                     // K-pair index into LDS
        v2f b0 = *(const v2f*)&Bp[kp][nl][0];
        v2f b1 = *(const v2f*)&Bp[kp][16 + nl][0];
        v2f b2 = *(const v2f*)&Bp[kp][32 + nl][0];
        c0 = __builtin_amdgcn_wmma_f32_16x16x4_f32(false, a, false, b0, (short)0, c0, false, false);
        c1 = __builtin_amdgcn_wmma_f32_16x16x4_f32(false, a, false, b1, (short)0, c1, false, false);
        c2 = __builtin_amdgcn_wmma_f32_16x16x4_f32(false, a, false, b2, (short)0, c2, false, false);
    }

    // Store C (+ fused bias). C/D layout: VGPR v -> row v + 8*kh, col = lane&15.
    float* __restrict__ orow = xpre + row0 * (long)G;
    const float bs0 = bsum[nl];
    const float bs1 = bsum[16 + nl];
    const float bs2 = bsum[32 + nl];
#pragma unroll
    for (int v = 0; v < 8; ++v) {
        const int mr = v + 8 * kh;
        float* pr = orow + (long)mr * G;
        pr[nl]      = c0[v] + bs0;
        pr[16 + nl] = c1[v] + bs1;
        if (32 + nl < G) pr[32 + nl] = c2[v] + bs2;
    }
}

// ---------------------------------------------------------------------------
// Phase 2: sequential scan. One thread per batch element (512 lanes = 16
// waves, spread as 16 single-wave blocks). h/c live in registers, W_hh in LDS
// (rows padded 9->12 floats for aligned ds_load_b128). Prefetch x_pre 8 steps
// ahead to hide HBM latency at low concurrency.
// ---------------------------------------------------------------------------
__device__ __forceinline__ float fexp(float v)  { return __builtin_amdgcn_exp2f(v * 1.4426950408889634f); }
__device__ __forceinline__ float fsig(float v)  { return __builtin_amdgcn_rcpf(1.0f + fexp(-v)); }
__device__ __forceinline__ float ftanh(float v) { return 1.0f - 2.0f * __builtin_amdgcn_rcpf(1.0f + fexp(2.0f * v)); }

__global__ __launch_bounds__(32) void lstm_scan(
    const float* __restrict__ xpre,                 // null in fused mode
    const float* __restrict__ h0, const float* __restrict__ c0,
    const float* __restrict__ W_hh, float* __restrict__ out,
    int fused,
    const float* __restrict__ x, const float* __restrict__ W_ih,
    const float* __restrict__ b_ih, const float* __restrict__ b_hh)
{
    __shared__ alignas(16) float wlds[G][12];       // W_hh rows, zero-padded
    __shared__ alignas(16) float wih[NIN][G];       // W_ih^T (fused mode only)
    __shared__ float bs[G];

    const int tid = threadIdx.x;
    for (int idx = tid; idx < G * 12; idx += 32) {
        int j = idx / 12, q = idx % 12;
        wlds[j][q] = (q < NH) ? W_hh[j * NH + q] : 0.0f;
    }
    if (fused) {
        for (int idx = tid; idx < NIN * G; idx += 32) {
            int k = idx / G, j = idx % G;
            wih[k][j] = W_ih[j * NIN + k];
        }
        for (int j = tid; j < G; j += 32) bs[j] = b_ih[j] + b_hh[j];
    }
    __syncthreads();

    const int b = blockIdx.x * 32 + tid;            // batch element [0,512)
    float h[NH], c[NH];
#pragma unroll
    for (int j = 0; j < NH; ++j) { h[j] = h0[b * NH + j]; c[j] = c0[b * NH + j]; }

    const float* xp   = fused ? nullptr : (xpre + (size_t)b * G);
    const float* xrow = fused ? (x + (size_t)b * NIN) : nullptr;
    float* orow = out + (size_t)b * NH;
    const size_t xp_stride = (size_t)BB * G;
    const size_t x_stride  = (size_t)BB * NIN;
    const size_t o_stride  = (size_t)BB * NH;

    for (int t = 0; t < TT; ++t) {
        float acc[G];
        if (!fused) {
            if (t + 8 < TT) {                       // hide HBM latency
                const char* pf = (const char*)(xp + 8 * xp_stride);
                __builtin_prefetch(pf, 0, 1);
                __builtin_prefetch(pf + 64, 0, 1);
                __builtin_prefetch(pf + 128, 0, 1);
            }
            const float4* xr4 = (const float4*)xp;  // 144B rows, 16B-aligned
#pragma unroll
            for (int q = 0; q < G / 4; ++q) {
                float4 v = xr4[q];
                acc[q * 4 + 0] = v.x; acc[q * 4 + 1] = v.y;
                acc[q * 4 + 2] = v.z; acc[q * 4 + 3] = v.w;
            }
            xp += xp_stride;
        } else {                                    // inline projection fallback
#pragma unroll
            for (int j = 0; j < G; ++j) acc[j] = bs[j];
            for (int k = 0; k < NIN; ++k) {
                float xk = xrow[k];
#pragma unroll
                for (int j = 0; j < G; ++j) acc[j] += xk * wih[k][j];
            }
            xrow += x_stride;
        }

        // gates += h @ W_hh^T   (2 x ds_load_b128 + 1 ds_load_b32 + 9 FMA per gate)
#pragma unroll
        for (int j = 0; j < G; ++j) {
            const float4* wr = (const float4*)&wlds[j][0];
            float4 a0 = wr[0], a1 = wr[1];
            float  w8 = wlds[j][8];
            acc[j] += a0.x * h[0] + a0.y * h[1] + a0.z * h[2] + a0.w * h[3]
                    + a1.x * h[4] + a1.y * h[5] + a1.z * h[6] + a1.w * h[7]
                    + w8 * h[8];
        }

#pragma unroll
        for (int j = 0; j < NH; ++j) {              // PyTorch gate order i,f,g,o
            float ig = fsig(acc[j]);
            float fg = fsig(acc[NH + j]);
            float gg = ftanh(acc[2 * NH + j]);
            float og = fsig(acc[3 * NH + j]);
            float cn = fg * c[j] + ig * gg;
            float hn = og * ftanh(cn);
            c[j] = cn; h[j] = hn;
            orow[j] = hn;
        }
        orow += o_stride;
    }

    // final states: hn then cn, appended after the (T,B,NH) output
    float* hnp = out + (size_t)TT * BB * NH + (size_t)b * NH;
    float* cnp = hnp + (size_t)BB * NH;
#pragma unroll
    for (int j = 0; j < NH; ++j) { hnp[j] = h[j]; cnp[j] = c[j]; }
}

// ---------------------------------------------------------------------------
extern "C" void kernel_launch(void* const* d_in, const int* in_sizes, int n_in,
                              void* d_out, int out_size, void* d_ws, size_t ws_size,
                              hipStream_t stream) {
    const float* x    = (const float*)d_in[0];      // (T,B,NIN)
    const float* h0   = (const float*)d_in[1];      // (1,B,NH)
    const float* c0   = (const float*)d_in[2];      // (1,B,NH)
    const float* W_ih = (const float*)d_in[3];      // (4H,NIN)
    const float* W_hh = (const float*)d_in[4];      // (4H,NH)
    const float* b_ih = (const float*)d_in[5];      // (4H,)
    const float* b_hh = (const float*)d_in[6];      // (4H,)
    float* out = (float*)d_out;

    const size_t need = (size_t)TT * BB * G * sizeof(float);   // ~302 MB
    if (ws_size >= need) {
        float* xpre = (float*)d_ws;
        // 131072 M-tiles, 8 waves (tiles) per block
        lstm_xproj_wmma<<<16384, 256, 0, stream>>>(x, W_ih, b_ih, b_hh, xpre);
        lstm_scan<<<16, 32, 0, stream>>>(xpre, h0, c0, W_hh, out, 0,
                                         nullptr, nullptr, nullptr, nullptr);
    } else {
        // workspace too small: fused fallback (projection recomputed in-scan)
        lstm_scan<<<16, 32, 0, stream>>>(nullptr, h0, c0, W_hh, out, 1,
                                         x, W_ih, b_ih, b_hh);
    }
}